// MaxPoolNd_39573828665510
// MI455X (gfx1250) — compile-verified
//
#include <hip/hip_runtime.h>
#include <cstdint>

// out[n, a, c', b', d', e'] = max over 2x2x2x2 window in (c, b, d, e) of
//                            x[n, 2c'+cc, a, 2b'+bb, 2d'+dd, 2e'+ee]
// x: (2, 32, 32, 32, 32, 32) f32, out: (2, 32, 16, 16, 16, 16) f32.
//
// One block per (n, a, c', b'): stages the four 4KB (d,e) input planes into
// LDS with CDNA5 async global->LDS loads (ASYNCcnt), then each of 256 threads
// reduces its 16-element window and writes one output (1KB contiguous/block).

#define LDS_AS __attribute__((address_space(3)))

__global__ __launch_bounds__(256) void maxpool4d_async_kernel(
    const float* __restrict__ x, float* __restrict__ out) {
  __shared__ float smem[4 * 1024];  // 4 planes x (32x32 f32) = 16 KB

  const unsigned tid = threadIdx.x;
  const unsigned blk = blockIdx.x;
  const unsigned bp = blk & 15u;          // b'
  const unsigned cp = (blk >> 4) & 15u;   // c'
  const unsigned a  = (blk >> 8) & 31u;   // a
  const unsigned n  = blk >> 13;          // n

  // Raw LDS byte offset of smem (addrspace(3) pointers are 32-bit offsets).
  const unsigned lds_base = (unsigned)(uintptr_t)(LDS_AS float*)smem;

  // Stage 4 planes: plane p <-> (cc = p>>1, bb = p&1). Each lane moves 16 B,
  // 256 threads cover one contiguous 4 KB plane per instruction.
#pragma unroll
  for (int p = 0; p < 4; ++p) {
    const unsigned c = 2u * cp + (unsigned)(p >> 1);
    const unsigned b = 2u * bp + (unsigned)(p & 1);
    // element offset of plane start: ((((n*32 + c)*32 + a)*32 + b) * 1024
    const uint64_t plane_elt =
        ((((uint64_t)n * 32u + c) * 32u + a) * 32u + b) * 1024u;
    const uint64_t gaddr =
        (uint64_t)(uintptr_t)x + plane_elt * 4u + (uint64_t)tid * 16u;
    const unsigned laddr = lds_base + (unsigned)p * 4096u + tid * 16u;
    asm volatile("global_load_async_to_lds_b128 %0, %1, off"
                 :: "v"(laddr), "v"(gaddr)
                 : "memory");
  }
  // Wait for this wave's async LDS fills, then sync all 8 waves of the block.
  asm volatile("s_wait_asynccnt 0x0" ::: "memory");
  __syncthreads();

  const unsigned ep = tid & 15u;   // e'
  const unsigned dp = tid >> 4;    // d'

  float m = -__builtin_inff();
#pragma unroll
  for (int p = 0; p < 4; ++p) {
    const float2* pl = (const float2*)(smem + p * 1024);
    const unsigned i0 = dp * 32u + ep;        // row d=2d', e-pair at 2e'
    const float2 t0 = pl[i0];                 // ds_load_b64
    const float2 t1 = pl[i0 + 16u];           // row d=2d'+1
    m = fmaxf(m, fmaxf(fmaxf(t0.x, t0.y), fmaxf(t1.x, t1.y)));
  }

  // out flat index = blk*256 + (d'*16 + e') : contiguous, coalesced store.
  out[(uint64_t)blk * 256u + tid] = m;
}

extern "C" void kernel_launch(void* const* d_in, const int* in_sizes, int n_in,
                              void* d_out, int out_size, void* d_ws, size_t ws_size,
                              hipStream_t stream) {
  const float* x = (const float*)d_in[0];   // (2,32,32,32,32,32) f32
  float* out = (float*)d_out;               // (2,32,16,16,16,16) f32
  (void)in_sizes; (void)n_in; (void)out_size; (void)d_ws; (void)ws_size;

  const unsigned blocks = 2u * 32u * 16u * 16u;  // (n, a, c', b') = 16384
  maxpool4d_async_kernel<<<blocks, 256, 0, stream>>>(x, out);
}